// NN_CYK_Model_38268158608147
// MI455X (gfx1250) — compile-verified
//
#include <hip/hip_runtime.h>
#include <hip/hip_bf16.h>
#include <math.h>

// Problem constants (from setup_inputs): N=16, V=100000, E=512, T=64, NT=32,
// S=96, s_dim=256, BEAM=8.
#define CYK_N    16
#define CYK_V    100000
#define CYK_E    512
#define CYK_T    64
#define CYK_NT   32
#define CYK_S    96
#define CYK_D    256   // s_dim
#define CYK_B    8     // beam

typedef __attribute__((ext_vector_type(2))) float v2f;
typedef __attribute__((ext_vector_type(8))) float v8f;

// Offset (in beam-slot units; x256 for feature vectors) of chart level l
// (1-based). Level l has (17-l) cells, 8 beams each.
__host__ __device__ __forceinline__ int sc_off(int l) {
    return 8 * ((l - 1) * 17 - ((l - 1) * l) / 2);
}

// ---------------------------------------------------------------------------
// 256x256 transpose (LDS tiled): out[n*256+k] = in[k*256+n].
// grid = 256 blocks of 256 threads; block handles one 16x16 tile.
// ---------------------------------------------------------------------------
__global__ void cyk_transpose_kernel(const float* __restrict__ in,
                                     float* __restrict__ out) {
    __shared__ float tile[16][17];
    const int bx = blockIdx.x & 15;        // column tile of `in`
    const int by = blockIdx.x >> 4;        // row tile of `in`
    const int tx = threadIdx.x & 15;
    const int ty = threadIdx.x >> 4;
    tile[ty][tx] = in[(by * 16 + ty) * CYK_D + bx * 16 + tx];
    __syncthreads();
    out[(bx * 16 + ty) * CYK_D + by * 16 + tx] = tile[tx][ty];
}

// ---------------------------------------------------------------------------
// Level-1: feat0 = tanh(emb @ Wp_word + pret @ Wp_gram); per-word top-8 of the
// 64 preterminal scores. One block (256 threads) per word.
// ---------------------------------------------------------------------------
__global__ void cyk_level1_kernel(const int* __restrict__ word_ids,
                                  const float* __restrict__ emb_tab,
                                  const float* __restrict__ gp,      // (T,V)
                                  const float* __restrict__ Wp_word, // (E,D)
                                  const float* __restrict__ Wp_gram, // (T,D)
                                  float* __restrict__ feat_pool,
                                  float* __restrict__ sc_pool,
                                  int* __restrict__ sym_pool) {
    const int n = blockIdx.x;      // word index 0..15
    const int d = threadIdx.x;     // feature dim 0..255
    const int wid = word_ids[n];

    __shared__ float s_pret[CYK_T];
    __shared__ float s_emb[CYK_E];
    if (d < CYK_T) s_pret[d] = gp[(long)d * CYK_V + wid];
    const float* e = emb_tab + (long)wid * CYK_E;
    s_emb[d]       = e[d];
    s_emb[d + 256] = e[d + 256];
    __syncthreads();

    float sum = 0.f;
    #pragma unroll 8
    for (int k = 0; k < CYK_E; ++k) sum += s_emb[k] * Wp_word[k * CYK_D + d];
    #pragma unroll 8
    for (int t = 0; t < CYK_T; ++t) sum += s_pret[t] * Wp_gram[t * CYK_D + d];
    const float f = tanhf(sum);
    for (int b = 0; b < CYK_B; ++b)
        feat_pool[((long)n * CYK_B + b) * CYK_D + d] = f;

    // top-8 of 64 (descending, ties -> lower index): serial on thread 0.
    if (d == 0) {
        unsigned long long used = 0ull;
        for (int k = 0; k < CYK_B; ++k) {
            float best = -3.402823466e38f; int bi = 0;
            for (int t = 0; t < CYK_T; ++t) {
                if ((used >> t) & 1ull) continue;
                if (s_pret[t] > best) { best = s_pret[t]; bi = t; }
            }
            used |= (1ull << bi);
            sc_pool[n * CYK_B + k] = best;
            sym_pool[n * CYK_B + k] = CYK_NT + bi;
        }
    }
}

// ---------------------------------------------------------------------------
// Projection GEMM via f32 WMMA: out[R,256] = feats[R,256] @ W[256,256] for
// W = Wm_left and W = Wm_right (both supplied pre-TRANSPOSED as Wt[n][k] so
// the B-fragment load is contiguous). One wave32 per 16x16 output tile,
// K stepped by 4 (v_wmma_f32_16x16x4_f32). Row index is CLAMPED (not
// predicated) for the A ragged edge: out-of-range A rows only influence
// out-of-range D rows, whose stores are guarded — keeps EXEC uniform, no
// divergent loads. gridDim.x = Mtiles*16*2, blockDim.x = 32.
// ---------------------------------------------------------------------------
__global__ void cyk_project_kernel(const float* __restrict__ feats,
                                   const float* __restrict__ WlT,  // (D,D) transposed
                                   const float* __restrict__ WrT,  // (D,D) transposed
                                   float* __restrict__ outL,
                                   float* __restrict__ outR,
                                   int R) {
    int bid = blockIdx.x;
    const int w  = bid & 1;  bid >>= 1;   // 0 -> left, 1 -> right
    const int nt = bid & 15; bid >>= 4;   // N tile (16 of them)
    const int mt = bid;                   // M tile

    const float* Wt = w ? WrT  : WlT;
    float*      out = w ? outR : outL;

    const int lane = threadIdx.x;         // 0..31 (wave32)
    const int half = lane >> 4;           // lanes 16-31 carry K+2/K+3
    const int l16  = lane & 15;

    int row_a = mt * 16 + l16;            // A-matrix row (M) for this lane
    if (row_a >= R) row_a = R - 1;        // clamp: no divergence, writes guarded
    const int col_b = nt * 16 + l16;      // B-matrix column (N) for this lane

    const float* arow = feats + (long)row_a * CYK_D + 2 * half;
    const float* brow = Wt    + (long)col_b * CYK_D + 2 * half;

    v8f acc = {};
    #pragma unroll 8
    for (int k0 = 0; k0 < CYK_D; k0 += 4) {
        const v2f a = *(const v2f*)(arow + k0);   // global_load_b64
        const v2f b = *(const v2f*)(brow + k0);   // global_load_b64
        // D = A(16x4,f32) x B(4x16,f32) + C  ->  v_wmma_f32_16x16x4_f32
        acc = __builtin_amdgcn_wmma_f32_16x16x4_f32(
            /*neg_a=*/false, a, /*neg_b=*/false, b,
            /*c_mod=*/(short)0, acc, /*reuse_a=*/false, /*reuse_b=*/false);
    }

    // D layout: lanes 0-15 -> M = v, lanes 16-31 -> M = v + 8 ; N = l16.
    #pragma unroll
    for (int v = 0; v < 8; ++v) {
        const int m = mt * 16 + v + 8 * half;
        if (m < R) out[(long)m * CYK_D + nt * 16 + l16] = acc[v];
    }
}

// ---------------------------------------------------------------------------
// Candidate scores for level L. grid = dim3((L-1)*10, C); one thread per
// candidate, so the div/mod is by the compile-time constant 2560.
// Column layout per split s (1..L-1): 2048 pair scores (i*8+j)*32+n, then
// 512 unary scores a*32+n — matching the reference's concatenation order.
// ---------------------------------------------------------------------------
__global__ void cyk_score_kernel(const float* __restrict__ sc_pool,
                                 const int* __restrict__ sym_pool,
                                 const float* __restrict__ G,   // (NT,S,S)
                                 const float* __restrict__ GS,  // (S,NT)
                                 float* __restrict__ scores,
                                 int L) {
    const int c   = blockIdx.y;                       // cell
    const int col = blockIdx.x * 256 + threadIdx.x;   // candidate column
    const int s = col / 2560 + 1;                     // split (const divisor)
    const int r = col % 2560;

    const float* scs = sc_pool  + sc_off(s);
    const int*   sys = sym_pool + sc_off(s);
    const float* scr = sc_pool  + sc_off(L - s);
    const int*   syr = sym_pool + sc_off(L - s);

    float val;
    if (r < 2048) {
        const int i = r >> 8, j = (r >> 5) & 7, n = r & 31;
        const int lsym = sys[c * CYK_B + i];
        const int rsym = syr[(s + c) * CYK_B + j];
        val = scs[c * CYK_B + i] * scr[(s + c) * CYK_B + j] *
              G[((long)n * CYK_S + lsym) * CYK_S + rsym];
    } else {
        const int r2 = r - 2048, a = r2 >> 5, n = r2 & 31;
        float v; int sym;
        if (a < CYK_B) { v = scs[c * CYK_B + a];             sym = sys[c * CYK_B + a]; }
        else           { v = scr[(s + c) * CYK_B + (a - 8)]; sym = syr[(s + c) * CYK_B + (a - 8)]; }
        val = v * GS[sym * CYK_NT + n];
    }
    scores[(long)c * ((L - 1) * 2560) + col] = val;
}

// ---------------------------------------------------------------------------
// Per-cell top-8 (jax top_k semantics: descending, ties -> lower index) and
// reconstruction of only the 8 winning feature vectors from the projected
// charts. One block (256 threads == feature dim) per cell.
// ---------------------------------------------------------------------------
__global__ void cyk_topk_kernel(const float* __restrict__ scores,
                                const float* __restrict__ projL_pool,
                                const float* __restrict__ projR_pool,
                                float* __restrict__ sc_pool,
                                int* __restrict__ sym_pool,
                                float* __restrict__ feat_pool,
                                int L,
                                float* __restrict__ d_out /* non-null at L==16 */) {
    const int c = blockIdx.x;
    const int M = (L - 1) * 2560;
    const float* srow = scores + (long)c * M;
    const int tid = threadIdx.x;

    __shared__ float s_best[256];
    __shared__ int   s_bidx[256];
    __shared__ int   chosen[CYK_B];

    for (int k = 0; k < CYK_B; ++k) {
        float best = -3.402823466e38f;
        int   bidx = 0x7fffffff;
        for (int t = tid; t < M; t += 256) {
            bool taken = false;
            for (int q = 0; q < k; ++q) taken |= (chosen[q] == t);
            if (taken) continue;
            const float v = srow[t];
            if (v > best || (v == best && t < bidx)) { best = v; bidx = t; }
        }
        s_best[tid] = best; s_bidx[tid] = bidx;
        __syncthreads();
        for (int off = 128; off > 0; off >>= 1) {
            if (tid < off) {
                const float v2 = s_best[tid + off];
                const int   i2 = s_bidx[tid + off];
                if (v2 > s_best[tid] || (v2 == s_best[tid] && i2 < s_bidx[tid])) {
                    s_best[tid] = v2; s_bidx[tid] = i2;
                }
            }
            __syncthreads();
        }
        if (tid == 0) {
            chosen[k] = s_bidx[0];
            sc_pool[sc_off(L) + c * CYK_B + k] = s_best[0];
        }
        __syncthreads();
    }

    // Materialize only the 8 winning feature vectors.
    const int d = tid;   // exactly 256 threads == feature dims
    for (int k = 0; k < CYK_B; ++k) {
        const int t = chosen[k];
        const int s = t / 2560 + 1;
        const int r = t % 2560;
        float fv; int n;
        if (r < 2048) {
            const int i = r >> 8, j = (r >> 5) & 7; n = r & 31;
            const float* pl = projL_pool + (long)(sc_off(s) + c * CYK_B + i) * CYK_D;
            const float* pr = projR_pool + (long)(sc_off(L - s) + (s + c) * CYK_B + j) * CYK_D;
            fv = tanhf(pl[d] + pr[d]);
        } else {
            const int r2 = r - 2048, a = r2 >> 5; n = r2 & 31;
            const float* p = (a < CYK_B)
                ? projL_pool + (long)(sc_off(s) + c * CYK_B + a) * CYK_D
                : projL_pool + (long)(sc_off(L - s) + (s + c) * CYK_B + (a - 8)) * CYK_D;
            fv = tanhf(p[d]);
        }
        feat_pool[(long)(sc_off(L) + c * CYK_B + k) * CYK_D + d] = fv;
        if (d == 0) sym_pool[sc_off(L) + c * CYK_B + k] = n;
        if (d_out != nullptr && c == 0 && k == 0) d_out[d] = fv;  // chart[16][1][0,0]
    }
}

// ---------------------------------------------------------------------------
extern "C" void kernel_launch(void* const* d_in, const int* in_sizes, int n_in,
                              void* d_out, int out_size, void* d_ws, size_t ws_size,
                              hipStream_t stream) {
    (void)in_sizes; (void)n_in; (void)out_size; (void)ws_size;

    const int*   word_ids = (const int*)  d_in[0];
    const float* emb_tab  = (const float*)d_in[1];   // (V,E)
    const float* gp       = (const float*)d_in[2];   // (T,V)
    const float* G        = (const float*)d_in[3];   // (NT,S,S)
    const float* GS       = (const float*)d_in[4];   // (S,NT)
    const float* Wp_word  = (const float*)d_in[5];   // (E,D)
    const float* Wp_gram  = (const float*)d_in[6];   // (T,D)
    const float* Wm_left  = (const float*)d_in[7];   // (D,D)
    const float* Wm_right = (const float*)d_in[8];   // (D,D)

    // Workspace carve-up (floats). Cells*beams total = 1088; feature pools
    // 1088*256 = 278528 floats each; max score buffer 163840 floats;
    // two transposed 256x256 weights. Total ~4.5 MB (all L2-resident).
    float* ws         = (float*)d_ws;
    float* feat_pool  = ws;
    float* projL_pool = feat_pool  + 278528;
    float* projR_pool = projL_pool + 278528;
    float* sc_pool    = projR_pool + 278528;              // 1088 floats
    int*   sym_pool   = (int*)(sc_pool + 1088);           // 1088 ints
    float* scores     = (float*)(sym_pool + 1088);        // <= 163840 floats
    float* WlT        = scores + 163840;                  // 65536 floats
    float* WrT        = WlT    + 65536;                   // 65536 floats

    // One-time (per launch) weight transposes for contiguous WMMA B-loads.
    cyk_transpose_kernel<<<256, 256, 0, stream>>>(Wm_left,  WlT);
    cyk_transpose_kernel<<<256, 256, 0, stream>>>(Wm_right, WrT);

    // Level 1: leaf features + per-word beam.
    cyk_level1_kernel<<<CYK_N, 256, 0, stream>>>(
        word_ids, emb_tab, gp, Wp_word, Wp_gram, feat_pool, sc_pool, sym_pool);

    // Project level-1 features with both weight matrices (WMMA).
    {
        const int R = CYK_N * CYK_B;                      // 128 rows
        const int mtiles = (R + 15) / 16;
        cyk_project_kernel<<<mtiles * 16 * 2, 32, 0, stream>>>(
            feat_pool, WlT, WrT, projL_pool, projR_pool, R);
    }

    for (int L = 2; L <= CYK_N; ++L) {
        const int C = CYK_N + 1 - L;

        // Scores: one thread per candidate; 2560 = 10 blocks of 256 per split.
        cyk_score_kernel<<<dim3((L - 1) * 10, C), 256, 0, stream>>>(
            sc_pool, sym_pool, G, GS, scores, L);

        // Top-8 per cell + winning features (+ final output at L==16).
        cyk_topk_kernel<<<C, 256, 0, stream>>>(
            scores, projL_pool, projR_pool, sc_pool, sym_pool, feat_pool,
            L, (L == CYK_N) ? (float*)d_out : nullptr);

        // Project this level's new features for use as children later.
        if (L < CYK_N) {
            const int R = C * CYK_B;
            const int mtiles = (R + 15) / 16;
            const long base = (long)sc_off(L);
            cyk_project_kernel<<<mtiles * 16 * 2, 32, 0, stream>>>(
                feat_pool + base * CYK_D, WlT, WrT,
                projL_pool + base * CYK_D, projR_pool + base * CYK_D, R);
        }
    }
}